// Convolutional_56358560858517
// MI455X (gfx1250) — compile-verified
//
#include <hip/hip_runtime.h>

typedef __attribute__((ext_vector_type(2))) float v2f;
typedef __attribute__((ext_vector_type(8))) float v8f;

#define CIN   64
#define COUT  128
#define HH    112
#define WW    112
#define NPIX  (HH * WW)          // 12544
#define NBAT  32
#define KTOT  (CIN * 9)          // 576
#define KSLAB 16
#define NBLK  128                // spatial columns per workgroup
#define AKP   (KSLAB + 4)        // padded K stride (20 floats): conflict-free + 8B aligned

// CDNA5 async direct-to-LDS copy (GLOBAL_LOAD_ASYNC_TO_LDS_B32, ASYNCcnt).
__device__ __forceinline__ void async_copy_b32(unsigned lds_off, const void* gptr) {
    asm volatile("global_load_async_to_lds_b32 %0, %1, off"
                 :: "v"(lds_off), "v"((unsigned long long)gptr)
                 : "memory");
}
__device__ __forceinline__ void wait_async_all() {
    asm volatile("s_wait_asynccnt 0x0" ::: "memory");
}

__launch_bounds__(256, 1)
__global__ void conv3x3_wmma_f32(const float* __restrict__ x,
                                 const float* __restrict__ w,
                                 float* __restrict__ out,
                                 const float* __restrict__ zerow) {
    // Double-buffered K-slabs, transposed layout: fragment = contiguous {K,K+1} pair.
    __shared__ float Alds[2][COUT][AKP];   // weights,   [cout][k]   (20 KB)
    __shared__ float Blds[2][NBLK][AKP];   // im2col x,  [col][k]    (20 KB)

    const int tid   = threadIdx.x;
    const int lane  = tid & 31;
    const int wave  = tid >> 5;
    const int pbase = blockIdx.x * NBLK;   // first spatial column of this block
    const int mbase = wave * 16;           // this wave's Cout rows

    // ---- per-thread staging coordinates (loop-invariant) ----
    // A: 128x16 = 2048 elems, 8/thread. idx = tid + i*256; m = idx>>4, kk = idx&15.
    int a_m[8], a_kk[8];
#pragma unroll
    for (int i = 0; i < 8; ++i) {
        int idx = tid + i * 256;
        a_m[i]  = idx >> 4;
        a_kk[i] = idx & 15;
    }
    // B: 128x16 = 2048 elems, 8/thread. col = idx>>4 (0..127), kk = idx&15.
    int b_col[8], b_kk[8], b_n[8], b_h[8], b_w[8];
#pragma unroll
    for (int i = 0; i < 8; ++i) {
        int idx  = tid + i * 256;
        b_col[i] = idx >> 4;
        b_kk[i]  = idx & 15;
        int p    = pbase + b_col[i];
        int n    = p / NPIX;
        int rem  = p - n * NPIX;
        b_n[i]   = n;
        b_h[i]   = rem / WW;
        b_w[i]   = rem - (rem / WW) * WW;
    }

    auto stageA = [&](int buf, int ks) {
#pragma unroll
        for (int i = 0; i < 8; ++i) {
            int k   = ks + a_kk[i];
            int cin = k / 9;
            int r   = k - cin * 9;
            // W layout: (Cin, Cout, 3, 3) -> idx = cin*Cout*9 + m*9 + r
            async_copy_b32((unsigned)(unsigned long long)&Alds[buf][a_m[i]][a_kk[i]],
                           &w[cin * (COUT * 9) + a_m[i] * 9 + r]);
        }
    };
    auto stageB = [&](int buf, int ks) {
#pragma unroll
        for (int i = 0; i < 8; ++i) {
            int k   = ks + b_kk[i];
            int cin = k / 9;
            int r   = k - cin * 9;
            int u   = r / 3;
            int v   = r - u * 3;
            int hi  = b_h[i] + u - 1;
            int wi  = b_w[i] + v - 1;
            const float* src = zerow;          // OOB halo taps read a zeroed word
            if ((unsigned)hi < (unsigned)HH && (unsigned)wi < (unsigned)WW)
                src = &x[((b_n[i] * CIN + cin) * HH + hi) * WW + wi];
            async_copy_b32((unsigned)(unsigned long long)&Blds[buf][b_col[i]][b_kk[i]],
                           src);
        }
    };

    v8f acc[8] = {};   // 16(M) x 128(N) per wave: N-tiles at cols t*16 + coll

    stageA(0, 0);
    stageB(0, 0);
    wait_async_all();
    __syncthreads();

    // Fragment coordinates per ISA 7.12.2 (wave32):
    //  A 16x4: lanes 0-15 hold K={0,1}, lanes 16-31 hold K={2,3}; M = lane%16.
    //  B 4x16: same half-wave K split; N = lane%16.
    const int half = (lane < 16) ? 0 : 2;
    const int mrow = mbase + (lane & 15);
    const int coll = lane & 15;

    int buf = 0;
    for (int ks = 0; ks < KTOT; ks += KSLAB) {
        const int nbuf = buf ^ 1;
        if (ks + KSLAB < KTOT) {     // overlap next-slab async staging with compute
            stageA(nbuf, ks + KSLAB);
            stageB(nbuf, ks + KSLAB);
        }
#pragma unroll
        for (int k4 = 0; k4 < KSLAB; k4 += 4) {
            const int kf = k4 + half;
            v2f a = *(const v2f*)&Alds[buf][mrow][kf];
#pragma unroll
            for (int t = 0; t < 8; ++t) {
                v2f b = *(const v2f*)&Blds[buf][t * 16 + coll][kf];
                acc[t] = __builtin_amdgcn_wmma_f32_16x16x4_f32(
                    false, a, false, b, (short)0, acc[t], false, false);
            }
        }
        wait_async_all();            // async writes to LDS complete before publishing
        __syncthreads();
        buf = nbuf;
    }

    // ---- writeback: C/D layout: VGPR r -> M = r (lanes 0-15) / r+8 (lanes 16-31) ----
    const int mout = mbase + ((lane < 16) ? 0 : 8);
#pragma unroll
    for (int t = 0; t < 8; ++t) {
        int p   = pbase + t * 16 + coll;
        int n   = p / NPIX;
        int rem = p - n * NPIX;
        int h   = rem / WW;
        int wo  = rem - h * WW;
        float* o = &out[((n * COUT + mout) * HH + h) * WW + wo];
#pragma unroll
        for (int r = 0; r < 8; ++r) {
            o[r * (long)(HH * WW)] = acc[t][r];   // M stride = H*W in NCHW
        }
    }
}

extern "C" void kernel_launch(void* const* d_in, const int* in_sizes, int n_in,
                              void* d_out, int out_size, void* d_ws, size_t ws_size,
                              hipStream_t stream) {
    const float* x = (const float*)d_in[0];   // (32, 64, 112, 112) fp32
    const float* w = (const float*)d_in[1];   // (64, 128, 3, 3)    fp32
    float* out = (float*)d_out;               // (32, 128, 112, 112) fp32

    // Zero word region for im2col halo taps (capture-safe stream memset).
    hipMemsetAsync(d_ws, 0, 256, stream);

    const int total_cols = NBAT * NPIX;       // 401408
    const int grid = total_cols / NBLK;       // 3136 workgroups
    conv3x3_wmma_f32<<<grid, 256, 0, stream>>>(x, w, out, (const float*)d_ws);
}